// HybridFCLRegression_65481071397398
// MI455X (gfx1250) — compile-verified
//
#include <hip/hip_runtime.h>
#include <math.h>

typedef float v2f __attribute__((ext_vector_type(2)));
typedef float v8f __attribute__((ext_vector_type(8)));

#define NQ        16
#define DSTATE    65536            // 2^16 amplitudes per state
#define NPAIR     32768            // D/2 (qubit-0 pairs)
#define V4_HALF   8192             // NPAIR / 4 (float4 units per half-row)
#define BPR       16               // stage-1 blocks per batch row
#define V4_BLK    (V4_HALF / BPR)  // 512 float4 iterations per block
#define T1        256              // stage-1 block size (8 wave32)

// ---------------------------------------------------------------------------
// Stage 1: streaming quadratic-form partial sums.
// out[b] = cos^2 t * S1 - 2 sin t * S2 + 2 sin t cos t * S3, with
//   S1 = sum(|a0|^2 - |a1|^2),  S2 = sum(x0x1 + y0y1),  S3 = sum(x0y1 - y0x1)
// Each block reduces its chunk to one float in ws[b*BPR + blk].
// ---------------------------------------------------------------------------
__global__ __launch_bounds__(T1)
void zexp_stage1(const float* __restrict__ re,
                 const float* __restrict__ im,
                 const float* __restrict__ thetas,
                 float* __restrict__ ws)
{
    const int b   = blockIdx.y;
    const int blk = blockIdx.x;
    const int tid = threadIdx.x;

    const float4* re4 = (const float4*)(re + (size_t)b * DSTATE);
    const float4* im4 = (const float4*)(im + (size_t)b * DSTATE);

    float s1 = 0.f, s2 = 0.f, s3 = 0.f;
    const int start = blk * V4_BLK;

    #pragma unroll 2
    for (int v = start + tid; v < start + V4_BLK; v += T1) {
        float4 x0 = re4[v];
        float4 x1 = re4[v + V4_HALF];
        float4 y0 = im4[v];
        float4 y1 = im4[v + V4_HALF];

        s1 += (x0.x * x0.x + y0.x * y0.x) - (x1.x * x1.x + y1.x * y1.x);
        s1 += (x0.y * x0.y + y0.y * y0.y) - (x1.y * x1.y + y1.y * y1.y);
        s1 += (x0.z * x0.z + y0.z * y0.z) - (x1.z * x1.z + y1.z * y1.z);
        s1 += (x0.w * x0.w + y0.w * y0.w) - (x1.w * x1.w + y1.w * y1.w);

        s2 += x0.x * x1.x + y0.x * y1.x;
        s2 += x0.y * x1.y + y0.y * y1.y;
        s2 += x0.z * x1.z + y0.z * y1.z;
        s2 += x0.w * x1.w + y0.w * y1.w;

        s3 += x0.x * y1.x - y0.x * x1.x;
        s3 += x0.y * y1.y - y0.y * x1.y;
        s3 += x0.z * y1.z - y0.z * x1.z;
        s3 += x0.w * y1.w - y0.w * x1.w;
    }

    const float t  = thetas[0];
    const float cs = cosf(t);
    const float sn = sinf(t);
    float p = cs * cs * s1 - 2.f * sn * s2 + 2.f * sn * cs * s3;

    // wave32 butterfly reduction
    #pragma unroll
    for (int o = 16; o >= 1; o >>= 1)
        p += __shfl_xor(p, o, 32);

    __shared__ float lsum[T1 / 32];
    if ((tid & 31) == 0) lsum[tid >> 5] = p;
    __syncthreads();
    if (tid == 0) {
        float acc = 0.f;
        #pragma unroll
        for (int w = 0; w < T1 / 32; ++w) acc += lsum[w];
        ws[b * BPR + blk] = acc;
    }
}

// ---------------------------------------------------------------------------
// Stage 2: reduce ws (128 rows x BPR=16 partials) with f32 WMMA.
// D(16x16) = A(16x4) * B(4x16 ones) + C  => every column of row m holds the
// row-sum of A; accumulate 4 K-chunks to sum all 16 partials per batch.
// A layout (32-bit 16x4): lanes 0-15 M=0..15 {K=0,K=1}, lanes 16-31 {K=2,K=3}.
// D layout: VGPR j: lanes 0-15 -> M=j, lanes 16-31 -> M=j+8.
// One wave handles 16 batches; 8 waves cover B=128.
// ---------------------------------------------------------------------------
__global__ __launch_bounds__(256)
void zexp_stage2_wmma(const float* __restrict__ ws, float* __restrict__ out)
{
    const int tid  = threadIdx.x;
    const int lane = tid & 31;
    const int wv   = tid >> 5;            // 0..7
    const int base = wv * 16;             // batch tile base
    const int m    = lane & 15;           // A-matrix row
    const int kb   = (lane >> 4) << 1;    // K base within chunk: 0 or 2

    const float* row = ws + (size_t)(base + m) * BPR;

    v8f acc = {};
    v2f ones;
    ones.x = 1.f; ones.y = 1.f;

    #pragma unroll
    for (int c = 0; c < BPR / 4; ++c) {   // 4 accumulated WMMAs (K=16 total)
        v2f a;
        a.x = row[4 * c + kb];
        a.y = row[4 * c + kb + 1];
        acc = __builtin_amdgcn_wmma_f32_16x16x4_f32(
            /*neg_a=*/false, a, /*neg_b=*/false, ones,
            /*c_mod=*/(short)0, acc, /*reuse_a=*/false, /*reuse_b=*/false);
    }

    // Column N=0 of D lives in lane 0 (M=0..7) and lane 16 (M=8..15).
    if (lane == 0) {
        #pragma unroll
        for (int j = 0; j < 8; ++j) out[base + j] = acc[j];
    } else if (lane == 16) {
        #pragma unroll
        for (int j = 0; j < 8; ++j) out[base + 8 + j] = acc[j];
    }
}

extern "C" void kernel_launch(void* const* d_in, const int* in_sizes, int n_in,
                              void* d_out, int out_size, void* d_ws, size_t ws_size,
                              hipStream_t stream)
{
    const float* re     = (const float*)d_in[0];
    const float* im     = (const float*)d_in[1];
    const float* thetas = (const float*)d_in[2];
    float*       out    = (float*)d_out;
    float*       ws     = (float*)d_ws;

    const int B = in_sizes[0] / DSTATE;   // 128

    dim3 grid1(BPR, B);
    zexp_stage1<<<grid1, T1, 0, stream>>>(re, im, thetas, ws);

    // one block: 8 waves x 16 batches = 128 outputs
    zexp_stage2_wmma<<<1, 256, 0, stream>>>(ws, out);
}